// SlidedRelativePE_19224273617236
// MI455X (gfx1250) — compile-verified
//
#include <hip/hip_runtime.h>
#include <math.h>

#define BATCH 4
#define SEQ   2048
#define DIM   512
#define RADIUS 128
#define NREL  257          // 2R+1
#define QT    16           // query rows per block
#define WAVES 8
#define NTHREADS 256
#define QS_STRIDE  516     // 512 + 4 pad -> distinct LDS banks across M lanes
#define QREL_STRIDE 260    // 257 -> pad to 260 (stride%64==4)
#define KCHUNK 64
#define PT_STRIDE 68       // 64 + 4 pad

typedef float v2f __attribute__((ext_vector_type(2)));
typedef float v8f __attribute__((ext_vector_type(8)));

// D = A(16x4) * B(4x16) + C, all fp32, wave32.
// Lane layout (ISA 7.12.2): A: M = lane&15, VGPR0/1 hold K = (lane<16?{0,1}:{2,3});
// B: N = lane&15, same K split; C/D: N = lane&15, VGPRv holds M = v + (lane<16?0:8).
__device__ __forceinline__ v8f wmma_f32(v2f a, v2f b, v8f c) {
    return __builtin_amdgcn_wmma_f32_16x16x4_f32(false, a, false, b, (short)0, c,
                                                 false, false);
}

__global__ __launch_bounds__(NTHREADS)
void k1_energy_stats(const float* __restrict__ Q, const float* __restrict__ K,
                     const float* __restrict__ E, float* __restrict__ energy,
                     float* __restrict__ stats) {
    __shared__ float Qs[QT * QS_STRIDE];       // 33 KB
    __shared__ float Qrel[QT * QREL_STRIDE];   // 16.6 KB
    __shared__ float rowmax[QT];
    __shared__ float rowsum[QT];
    __shared__ float wmax[WAVES * QT];

    const int tid  = threadIdx.x;
    const int wid  = tid >> 5;
    const int lane = tid & 31;
    const int q0   = blockIdx.x * QT;
    const int b    = blockIdx.y;

    if (tid < QT) { rowmax[tid] = -3.0e38f; rowsum[tid] = 0.0f; }

    // ---- Phase A: Q tile -> LDS (float4 coalesced) ----
    {
        const float4* Qg = (const float4*)(Q + ((size_t)b * SEQ + q0) * DIM);
        for (int i = tid; i < QT * (DIM / 4); i += NTHREADS) {
            int m  = i >> 7;         // / (DIM/4)
            int c4 = i & 127;
            float4 v = Qg[(size_t)m * (DIM / 4) + c4];
            float* dst = &Qs[m * QS_STRIDE + c4 * 4];
            dst[0] = v.x; dst[1] = v.y; dst[2] = v.z; dst[3] = v.w;
        }
    }
    __syncthreads();

    const int n    = lane & 15;
    const int koff = (lane < 16) ? 0 : 2;
    const int Mb   = (lane < 16) ? 0 : 8;
    const float* Arow = &Qs[n * QS_STRIDE];

    // ---- Phase B: Q_rel = Q * rel_embeds^T via WMMA (17 N-tiles) ----
    for (int rt = wid; rt < (NREL + 15) / 16; rt += WAVES) {
        int r = rt * 16 + n;
        const float* Brow = E + (size_t)((r < NREL) ? r : (NREL - 1)) * DIM;
        v8f acc = {};
        #pragma unroll 4
        for (int d0 = 0; d0 < DIM; d0 += 4) {
            v2f a  = *(const v2f*)(Arow + d0 + koff);
            v2f bb = *(const v2f*)(Brow + d0 + koff);
            acc = wmma_f32(a, bb, acc);
        }
        if (r < NREL) {
            #pragma unroll
            for (int v = 0; v < 8; v++)
                Qrel[(Mb + v) * QREL_STRIDE + r] = acc[v];
        }
    }
    __syncthreads();

    // ---- Phase C: energy = (Q K^T + rel)/sqrt(d), track row max ----
    // Each wave computes a 16x64 slab (4 N-tiles) so one A fragment feeds
    // 4 back-to-back WMMAs per d-step.
    const float scale = 0.044194173824159216f;  // 1/sqrt(512)
    float vmax[8];
    #pragma unroll
    for (int v = 0; v < 8; v++) vmax[v] = -3.0e38f;

    for (int kg = wid; kg < SEQ / 64; kg += WAVES) {   // 32 slabs, 4 per wave
        const int kbase = kg * 64;
        const float* Krow = K + ((size_t)b * SEQ + kbase + n) * DIM;
        v8f acc[4] = {};
        #pragma unroll 4
        for (int d0 = 0; d0 < DIM; d0 += 4) {
            v2f a = *(const v2f*)(Arow + d0 + koff);
            #pragma unroll
            for (int t = 0; t < 4; t++) {
                v2f bb = *(const v2f*)(Krow + (size_t)t * 16 * DIM + d0 + koff);
                acc[t] = wmma_f32(a, bb, acc[t]);
            }
        }
        #pragma unroll
        for (int t = 0; t < 4; t++) {
            const int kb = kbase + t * 16;
            #pragma unroll
            for (int v = 0; v < 8; v++) {
                int m    = Mb + v;
                int diff = (kb + n) - (q0 + m);
                int r    = min(max(diff, -RADIUS), RADIUS) + RADIUS;
                float e  = (acc[t][v] + Qrel[m * QREL_STRIDE + r]) * scale;
                energy[((size_t)b * SEQ + q0 + m) * SEQ + kb + n] = e;
                vmax[v] = fmaxf(vmax[v], e);
            }
        }
    }
    // reduce row max across the 16 lanes sharing a row, publish per wave
    #pragma unroll
    for (int v = 0; v < 8; v++) {
        float x = vmax[v];
        x = fmaxf(x, __shfl_xor(x, 1, 16));
        x = fmaxf(x, __shfl_xor(x, 2, 16));
        x = fmaxf(x, __shfl_xor(x, 4, 16));
        x = fmaxf(x, __shfl_xor(x, 8, 16));
        if ((lane & 15) == 0) wmax[wid * QT + (Mb + v)] = x;
    }
    __threadfence();      // make energy stores visible before re-read
    __syncthreads();
    if (tid < QT) {
        float mx = wmax[tid];
        #pragma unroll
        for (int w = 1; w < WAVES; w++) mx = fmaxf(mx, wmax[w * QT + tid]);
        rowmax[tid] = mx;
    }
    __syncthreads();

    // ---- Phase D: row sum of exp(e - max) (re-read energies, L2-hot) ----
    {
        const int row = tid >> 4;
        const int sub = tid & 15;
        const float m = rowmax[row];
        const float* erow = energy + ((size_t)b * SEQ + q0 + row) * SEQ;
        float s = 0.0f;
        for (int j = sub; j < SEQ; j += 16) s += __expf(erow[j] - m);
        atomicAdd(&rowsum[row], s);
    }
    __syncthreads();
    if (tid < QT) {
        size_t idx = ((size_t)b * SEQ + q0 + tid) * 2;
        stats[idx + 0] = rowmax[tid];
        stats[idx + 1] = rowsum[tid];
    }
}

__global__ __launch_bounds__(NTHREADS)
void k2_softmax_pv(const float* __restrict__ V, float* __restrict__ attn,
                   const float* __restrict__ stats, float* __restrict__ Z) {
    __shared__ float Ptile[QT * PT_STRIDE];
    __shared__ float rm[QT];
    __shared__ float rs[QT];

    const int tid  = threadIdx.x;
    const int wid  = tid >> 5;
    const int lane = tid & 31;
    const int q0   = blockIdx.x * QT;
    const int b    = blockIdx.y;

    if (tid < QT) {
        size_t idx = ((size_t)b * SEQ + q0 + tid) * 2;
        rm[tid] = stats[idx + 0];
        rs[tid] = 1.0f / stats[idx + 1];
    }
    __syncthreads();

    const int n     = lane & 15;
    const int koff  = (lane < 16) ? 0 : 2;
    const int Mb    = (lane < 16) ? 0 : 8;
    const int nbase = wid * (DIM / WAVES);   // 64 Z columns per wave
    const float* Arow = &Ptile[n * PT_STRIDE];

    v8f acc[4] = {};

    for (int kc = 0; kc < SEQ; kc += KCHUNK) {
        // produce: softmax-normalize a 16x64 chunk, write final attn, stash in LDS
        for (int i = tid; i < QT * KCHUNK; i += NTHREADS) {
            int mrow = i >> 6;
            int cc   = i & (KCHUNK - 1);
            size_t g = ((size_t)b * SEQ + q0 + mrow) * SEQ + kc + cc;
            float p  = __expf(attn[g] - rm[mrow]) * rs[mrow];
            attn[g]  = p;
            Ptile[mrow * PT_STRIDE + cc] = p;
        }
        __syncthreads();

        // consume: acc += P(16x64) * V(64x64-slice) with fp32 WMMA
        #pragma unroll 4
        for (int kk = 0; kk < KCHUNK; kk += 4) {
            v2f a = *(const v2f*)(Arow + kk + koff);
            const float* Vrow0 = V + ((size_t)b * SEQ + kc + kk + koff) * DIM + nbase;
            const float* Vrow1 = Vrow0 + DIM;
            #pragma unroll
            for (int t = 0; t < 4; t++) {
                v2f bb;
                bb[0] = Vrow0[t * 16 + n];
                bb[1] = Vrow1[t * 16 + n];
                acc[t] = wmma_f32(a, bb, acc[t]);
            }
        }
        __syncthreads();
    }

    // write Z slice
    #pragma unroll
    for (int t = 0; t < 4; t++) {
        #pragma unroll
        for (int v = 0; v < 8; v++) {
            Z[((size_t)b * SEQ + q0 + Mb + v) * DIM + nbase + t * 16 + n] = acc[t][v];
        }
    }
}

extern "C" void kernel_launch(void* const* d_in, const int* in_sizes, int n_in,
                              void* d_out, int out_size, void* d_ws, size_t ws_size,
                              hipStream_t stream) {
    const float* Q = (const float*)d_in[0];
    const float* K = (const float*)d_in[1];
    const float* V = (const float*)d_in[2];
    const float* E = (const float*)d_in[3];

    float* attn  = (float*)d_out;                            // [B, Sq, Sk]
    float* Z     = attn + (size_t)BATCH * SEQ * SEQ;         // [B, Sq, D]
    float* stats = (float*)d_ws;                             // [B*Sq][2] = 64 KB

    dim3 grid(SEQ / QT, BATCH);
    k1_energy_stats<<<grid, NTHREADS, 0, stream>>>(Q, K, E, attn, stats);
    k2_softmax_pv<<<grid, NTHREADS, 0, stream>>>(V, attn, stats, Z);
}